// NeuralRenderer_29188597743688
// MI455X (gfx1250) — compile-verified
//
#include <hip/hip_runtime.h>

typedef __attribute__((ext_vector_type(2))) float v2f;
typedef __attribute__((ext_vector_type(8))) float v8f;
typedef int v4i __attribute__((vector_size(4 * sizeof(int))));

#if __has_builtin(__builtin_amdgcn_global_load_async_to_lds_b128)
#define HAVE_ASYNC_LDS 1
#endif

__device__ __forceinline__ void async_wait0() {
#ifdef HAVE_ASYNC_LDS
#if __has_builtin(__builtin_amdgcn_s_wait_asynccnt)
  __builtin_amdgcn_s_wait_asynccnt(0);
#else
  asm volatile("s_wait_asynccnt 0x0" ::: "memory");
#endif
#endif
}

// ---------------------------------------------------------------------------
// Stage 1: perspective projection of vertices to NDC xy.
// ---------------------------------------------------------------------------
__global__ void project_kernel(const float* __restrict__ verts,  // [B,V,3]
                               const float* __restrict__ cams,   // [B,3]
                               float* __restrict__ proj,         // [B,V,2]
                               int B, int V) {
  int tid = blockIdx.x * blockDim.x + threadIdx.x;
  if (tid >= B * V) return;
  int b = tid / V;
  float x = verts[(size_t)tid * 3 + 0];
  float y = verts[(size_t)tid * 3 + 1];
  float z = verts[(size_t)tid * 3 + 2];
  float f  = cams[b * 3 + 0];
  float cx = cams[b * 3 + 1];
  float cy = cams[b * 3 + 2];
  float inv_img = 1.0f / (cams[1] * 2.0f);   // cam[0,1]*2 per reference
  float invz = 1.0f / z;
  float px = f * x * invz + cx;
  float py = f * y * invz + cy;
  proj[(size_t)tid * 2 + 0] = px * inv_img * 2.0f - 1.0f;
  proj[(size_t)tid * 2 + 1] = py * inv_img * 2.0f - 1.0f;
}

// ---------------------------------------------------------------------------
// Stage 2: per-face edge coefficients, oriented so area > 0.
// Per face: [A0,B0,C0,0, A1,B1,C1,0, A2,B2,C2,0] (48 B).
// Degenerate / padded faces: A=B=0, C=-1 -> never inside.
// ---------------------------------------------------------------------------
__global__ void face_coef_kernel(const float* __restrict__ proj,   // [B,V,2]
                                 const int* __restrict__ faces,    // [B,F,3]
                                 float* __restrict__ coef,         // [B,FPAD,12]
                                 int B, int V, int F, int FPAD) {
  int tid = blockIdx.x * blockDim.x + threadIdx.x;
  if (tid >= B * FPAD) return;
  int b = tid / FPAD;
  int f = tid - b * FPAD;

  float e[12];
  bool ok = false;
  if (f < F) {
    size_t fb = ((size_t)b * F + f) * 3;
    int i0 = faces[fb + 0];
    int i1 = faces[fb + 1];
    int i2 = faces[fb + 2];
    const float* pb = proj + (size_t)b * V * 2;
    float v0x = pb[(size_t)i0 * 2], v0y = pb[(size_t)i0 * 2 + 1];
    float v1x = pb[(size_t)i1 * 2], v1y = pb[(size_t)i1 * 2 + 1];
    float v2x = pb[(size_t)i2 * 2], v2y = pb[(size_t)i2 * 2 + 1];
    float area = (v1x - v0x) * (v2y - v0y) - (v1y - v0y) * (v2x - v0x);
    if (fabsf(area) > 1e-12f) {
      ok = true;
      float s = area < 0.0f ? -1.0f : 1.0f;
      float A0 = s * (v2y - v1y), B0 = -s * (v2x - v1x);
      float C0 = -(A0 * v1x + B0 * v1y);
      float A1 = s * (v0y - v2y), B1 = -s * (v0x - v2x);
      float C1 = -(A1 * v2x + B1 * v2y);
      float A2 = s * (v1y - v0y), B2 = -s * (v1x - v0x);
      float C2 = -(A2 * v0x + B2 * v0y);
      e[0] = A0; e[1] = B0; e[2]  = C0; e[3]  = 0.0f;
      e[4] = A1; e[5] = B1; e[6]  = C1; e[7]  = 0.0f;
      e[8] = A2; e[9] = B2; e[10] = C2; e[11] = 0.0f;
    }
  }
  if (!ok) {
#pragma unroll
    for (int j = 0; j < 3; ++j) {
      e[j * 4 + 0] = 0.0f; e[j * 4 + 1] = 0.0f;
      e[j * 4 + 2] = -1.0f; e[j * 4 + 3] = 0.0f;
    }
  }
  float* o = coef + (size_t)tid * 12;
#pragma unroll
  for (int k = 0; k < 12; ++k) o[k] = e[k];
}

// ---------------------------------------------------------------------------
// Stage 3: WMMA rasterizer, double-buffered async LDS staging.
// Block = 8 waves = 128 consecutive pixels of one row; each wave owns 16 px.
// Window w of WIN face groups is prefetched (global_load_async_to_lds_b128)
// into LDS buffer (w&1) while window w-1 is consumed: 3x
// v_wmma_f32_16x16x4_f32 per group, acc[r] = max(acc[r], min3(w0,w1,w2)).
// Pixel covered iff any face's min-edge value >= 0.  Per-wave exact
// done-check per window; block-wide early exit via __syncthreads_and.
// ---------------------------------------------------------------------------
#define WIN 16

__global__ void __launch_bounds__(256)
raster_wmma_kernel(const float* __restrict__ coef,  // [B,FPAD,12]
                   float* __restrict__ out,         // [B,256,256]
                   int FG, int FPAD) {
  __shared__ __attribute__((aligned(16))) float lcoef[2][WIN * 192];  // 2x12KB

  int wave = threadIdx.x >> 5;
  int t = blockIdx.x * 8 + wave;     // wave id == 16-pixel tile id
  int lane = threadIdx.x & 31;
  int n = lane & 15;                 // face column (B/D) & pixel index (A)
  bool hi = lane >= 16;

  int b = t >> 12;                   // 4096 tiles per batch image
  int rem = t & 4095;
  int y = rem >> 4;
  int xBase = (rem & 15) << 4;

  float px = (2.0f * (float)(xBase + n) + 1.0f) * (1.0f / 256.0f) - 1.0f;
  float py = (2.0f * (float)y + 1.0f) * (1.0f / 256.0f) - 1.0f;

  // A-matrix, ISA 16x4 f32 layout: lanes 0-15 (K0,K1)=(px,py); 16-31 (1,0).
  v2f a;
  a.x = hi ? 1.0f : px;
  a.y = hi ? 0.0f : py;

  const float* coefB = coef + (size_t)b * FPAD * 12;
  v8f cz = {};
  float acc[8];
#pragma unroll
  for (int r = 0; r < 8; ++r) acc[r] = -1.0f;
  bool done = false;

  int nWin = (FG + WIN - 1) / WIN;

  // Stage a window of face-group coefficients into an LDS buffer (async).
  auto stage = [&](int w) {
    int baseG = w * WIN;
    int wCount = FG - baseG;
    if (wCount > WIN) wCount = WIN;
    int nVec = wCount * 48;                       // float4s in window
    const float* gwin = coefB + (size_t)baseG * 192;
    float* lbuf = &lcoef[w & 1][0];
    for (int i = threadIdx.x; i < nVec; i += 256) {
#ifdef HAVE_ASYNC_LDS
      __builtin_amdgcn_global_load_async_to_lds_b128(
          (__attribute__((address_space(1))) v4i*)(gwin + (size_t)i * 4),
          (__attribute__((address_space(3))) v4i*)(lbuf + i * 4),
          0, 0);
#else
      *(float4*)(lbuf + i * 4) = *(const float4*)(gwin + (size_t)i * 4);
#endif
    }
  };

  stage(0);  // prologue prefetch

  for (int w = 0; w < nWin; ++w) {
    async_wait0();      // this wave's loads for window w are complete
    __syncthreads();    // all waves' loads complete -> buffer (w&1) ready

    // Prefetch next window into the other buffer; overlaps with consume.
    // Safe: every wave passed the barrier above, so no wave is still
    // reading buffer ((w+1)&1) (its last readers were in iteration w-1).
    if (w + 1 < nWin) stage(w + 1);

    if (!done) {
      int baseG = w * WIN;
      int wCount = FG - baseG;
      if (wCount > WIN) wCount = WIN;
      const float* lbuf = &lcoef[w & 1][0];
      for (int g = 0; g < wCount; ++g) {
        const float* fb = lbuf + (g * 16 + n) * 12 + (hi ? 2 : 0);
        v2f b0 = *reinterpret_cast<const v2f*>(fb);
        v2f b1 = *reinterpret_cast<const v2f*>(fb + 4);
        v2f b2 = *reinterpret_cast<const v2f*>(fb + 8);

        v8f d0 = __builtin_amdgcn_wmma_f32_16x16x4_f32(false, a, false, b0,
                                                       (short)0, cz, false, false);
        v8f d1 = __builtin_amdgcn_wmma_f32_16x16x4_f32(false, a, false, b1,
                                                       (short)0, cz, false, false);
        v8f d2 = __builtin_amdgcn_wmma_f32_16x16x4_f32(false, a, false, b2,
                                                       (short)0, cz, false, false);
#pragma unroll
        for (int r = 0; r < 8; ++r) {
          float m = fminf(fminf(d0[r], d1[r]), d2[r]);   // v_min3_num_f32
          acc[r] = fmaxf(acc[r], m);
        }
      }
      // Exact per-wave coverage check (once per window).
      unsigned cov = 0;
#pragma unroll
      for (int r = 0; r < 8; ++r) {
        unsigned msk = __builtin_amdgcn_ballot_w32(acc[r] >= 0.0f);
        if (msk & 0xFFFFu) cov |= 1u << r;       // pixels M = r
        if (msk >> 16)     cov |= 1u << (r + 8); // pixels M = r + 8
      }
      done = (cov == 0xFFFFu);
    }
    // Barrier (protects LDS buffer reuse) + block-wide early exit.
    if (__syncthreads_and((int)done)) break;
  }

  // Final coverage from accumulators.
  unsigned cov = 0;
#pragma unroll
  for (int r = 0; r < 8; ++r) {
    unsigned msk = __builtin_amdgcn_ballot_w32(acc[r] >= 0.0f);
    if (msk & 0xFFFFu) cov |= 1u << r;
    if (msk >> 16)     cov |= 1u << (r + 8);
  }
  if (!hi) {
    out[((size_t)b << 16) + ((size_t)y << 8) + (size_t)(xBase + n)] =
        ((cov >> n) & 1u) ? 1.0f : 0.0f;
  }
}

// ---------------------------------------------------------------------------
extern "C" void kernel_launch(void* const* d_in, const int* in_sizes, int n_in,
                              void* d_out, int out_size, void* d_ws, size_t ws_size,
                              hipStream_t stream) {
  const float* verts = (const float*)d_in[0];
  const int*   faces = (const int*)d_in[1];
  const float* cams  = (const float*)d_in[2];
  float* out = (float*)d_out;

  int B = in_sizes[2] / 3;
  int V = in_sizes[0] / (3 * B);
  int F = in_sizes[1] / (3 * B);
  int FG = (F + 15) / 16;
  int FPAD = FG * 16;

  float* ws = (float*)d_ws;
  float* proj = ws;                                   // B*V*2 floats
  size_t projN = (size_t)B * V * 2;
  size_t coefOff = (projN + 3) & ~(size_t)3;          // 16-byte align
  float* coef = ws + coefOff;                         // B*FPAD*12 floats

  int t1 = B * V;
  project_kernel<<<(t1 + 255) / 256, 256, 0, stream>>>(verts, cams, proj, B, V);

  int t2 = B * FPAD;
  face_coef_kernel<<<(t2 + 255) / 256, 256, 0, stream>>>(proj, faces, coef,
                                                         B, V, F, FPAD);

  int tiles = B * 4096;  // B * (256*256 / 16) 16-pixel segments
  raster_wmma_kernel<<<tiles / 8, 256, 0, stream>>>(coef, out, FG, FPAD);
}